// HungarianMatcher_82952998354975
// MI455X (gfx1250) — compile-verified
//
#include <hip/hip_runtime.h>
#include <cstdint>
#include <cstddef>

// ---- problem constants (match reference) ----
#define BSZ    16
#define NQRY   900
#define NCLS   81
#define KDIM   80          // NC-1, GEMM K
#define T_EACH 50
#define T_TOT  800         // BS*T_EACH
#define NQ     14400       // BS*Q  (GEMM M)

typedef __attribute__((ext_vector_type(2))) float v2f;
typedef __attribute__((ext_vector_type(8))) float v8f;

// =====================================================================
// Kernel 1a: focal terms  d[n][c] = pos - neg ; split into d80[N][80] + dlast[N]
// =====================================================================
__global__ void focal_kernel(const float* __restrict__ logits,
                             float* __restrict__ d80,
                             float* __restrict__ dlast)
{
    int idx = blockIdx.x * blockDim.x + threadIdx.x;
    if (idx >= NQ * NCLS) return;
    int n = idx / NCLS;
    int c = idx - n * NCLS;
    float x    = logits[idx];
    float prob = 1.0f / (1.0f + __expf(-x));
    float om   = 1.0f - prob;
    float pos  = 0.25f * om * om     * (-__logf(prob + 1e-8f));
    float neg  = 0.75f * prob * prob * (-__logf(om   + 1e-8f));
    float d    = pos - neg;
    if (c < KDIM) d80[(size_t)n * KDIM + c] = d;
    else          dlast[n] = d;
}

// =====================================================================
// Kernel 1b: precompute xyxy corners + areas for pred and target boxes
//   (hoists per-pair-redundant VALU out of the 11.5M-pair tile kernel)
// =====================================================================
__global__ void box_prep_kernel(const float* __restrict__ pred_boxes,  // [NQ][4]
                                const float* __restrict__ tgt_boxes,   // [T_TOT][4]
                                float* __restrict__ pxyxy,             // [NQ][4]
                                float* __restrict__ parea,             // [NQ]
                                float* __restrict__ txyxy,             // [T_TOT][4]
                                float* __restrict__ tarea)             // [T_TOT]
{
    int idx = blockIdx.x * blockDim.x + threadIdx.x;
    if (idx < NQ) {
        const float4 b = ((const float4*)pred_boxes)[idx];
        float4 o;
        o.x = b.x - 0.5f * b.z; o.y = b.y - 0.5f * b.w;
        o.z = b.x + 0.5f * b.z; o.w = b.y + 0.5f * b.w;
        ((float4*)pxyxy)[idx] = o;
        parea[idx] = (o.z - o.x) * (o.w - o.y);
    } else if (idx < NQ + T_TOT) {
        const int t = idx - NQ;
        const float4 b = ((const float4*)tgt_boxes)[t];
        float4 o;
        o.x = b.x - 0.5f * b.z; o.y = b.y - 0.5f * b.w;
        o.z = b.x + 0.5f * b.z; o.w = b.y + 0.5f * b.w;
        ((float4*)txyxy)[t] = o;
        tarea[t] = (o.z - o.x) * (o.w - o.y);
    }
}

// =====================================================================
// Kernel 2: fused cost-tile kernel.  One wave -> one 16x16 tile of C.
//   cost_em via 20x V_WMMA_F32_16X16X4_F32 (K=80), then + class + L1 + GIoU.
// =====================================================================
__global__ void __launch_bounds__(256)
cost_tile_kernel(const float* __restrict__ d80,
                 const float* __restrict__ dlast,
                 const float* __restrict__ pred_boxes,  // [NQ][4] cxcywh
                 const float* __restrict__ pxyxy,       // [NQ][4] xyxy
                 const float* __restrict__ parea,       // [NQ]
                 const float* __restrict__ tgt_labels,  // [T_TOT][80]
                 const float* __restrict__ tgt_boxes,   // [T_TOT][4] cxcywh
                 const float* __restrict__ txyxy,       // [T_TOT][4] xyxy
                 const float* __restrict__ tarea,       // [T_TOT]
                 float* __restrict__ C)                 // [NQ][T_TOT]
{
    const int NT   = T_TOT / 16;                  // 50 tiles along targets
    const int wave = threadIdx.x >> 5;            // wave32
    const int lane = threadIdx.x & 31;
    int tile = blockIdx.x * (blockDim.x >> 5) + wave;
    if (tile >= (NQ / 16) * NT) return;           // wave-uniform (EXEC all-ones kept)

    const int tm = tile / NT, tn = tile - tm * NT;
    const int row0 = tm * 16, col0 = tn * 16;

    const int hi  = lane >> 4;                    // 0: K pair {0,1}, 1: K pair {2,3}
    const int l15 = lane & 15;

    // A frag source: d80[row0+l15][4s + 2*hi + {0,1}]   (aligned b64 loads)
    const float* Ap = d80        + (size_t)(row0 + l15) * KDIM + 2 * hi;
    // B frag source: tgt_labels[col0+l15][4s + 2*hi + {0,1}]
    const float* Bp = tgt_labels + (size_t)(col0 + l15) * KDIM + 2 * hi;

    v8f acc = {};
#pragma unroll
    for (int s = 0; s < KDIM / 4; ++s) {
        v2f a = *(const v2f*)(Ap + s * 4);
        v2f b = *(const v2f*)(Bp + s * 4);
        // D = A(16x4,f32) * B(4x16,f32) + C   -- full fp32 matrix core op
        acc = __builtin_amdgcn_wmma_f32_16x16x4_f32(
                  false, a, false, b, (short)0, acc, false, false);
    }

    // ---- per-lane column (target) data, loaded once ----
    const int col = col0 + l15;
    const float4 tb  = ((const float4*)tgt_boxes)[col];   // cxcywh (L1 term)
    const float4 txy = ((const float4*)txyxy)[col];       // xyxy
    const float  ta  = tarea[col];

    // class cost for this lane's 8 rows: two vector loads (16B-aligned)
    const float4 dlA = ((const float4*)(dlast + row0 + hi * 8))[0];
    const float4 dlB = ((const float4*)(dlast + row0 + hi * 8))[1];
    const float dl[8] = { dlA.x, dlA.y, dlA.z, dlA.w, dlB.x, dlB.y, dlB.z, dlB.w };

#pragma unroll
    for (int r = 0; r < 8; ++r) {
        const int row = row0 + r + hi * 8;
        const float4 pb  = ((const float4*)pred_boxes)[row];  // cxcywh
        const float4 pxy = ((const float4*)pxyxy)[row];       // xyxy
        const float  pa  = parea[row];

        // L1 bbox cost in cxcywh
        const float cbbox = fabsf(pb.x - tb.x) + fabsf(pb.y - tb.y)
                          + fabsf(pb.z - tb.z) + fabsf(pb.w - tb.w);
        // GIoU from precomputed corners/areas
        const float iw    = fmaxf(0.0f, fminf(pxy.z, txy.z) - fmaxf(pxy.x, txy.x));
        const float ih    = fmaxf(0.0f, fminf(pxy.w, txy.w) - fmaxf(pxy.y, txy.y));
        const float inter = iw * ih;
        const float uni   = pa + ta - inter;
        const float ew    = fmaxf(pxy.z, txy.z) - fminf(pxy.x, txy.x);  // >= 0 always
        const float eh    = fmaxf(pxy.w, txy.w) - fminf(pxy.y, txy.y);
        const float ea    = ew * eh;
        // giou = inter/uni - (ea-uni)/ea ; fast rcp-based divide (v_rcp co-issues)
        const float giou  = __fdividef(inter, uni) - __fdividef(ea - uni, ea);

        const float out = acc[r] * (1.0f / (float)KDIM)   // cost_em
                        + dl[r]                           // cost_class
                        + cbbox                           // cost_bbox
                        - giou;                           // cost_giou
        C[(size_t)row * T_TOT + col] = out;
    }
}

// =====================================================================
// Kernel 3: Jonker-Volgenant LSA, one workgroup per batch.
//   ct = cost.T : n=50 rows, m=900 cols; inner O(m) scans parallelized.
// =====================================================================
#define LSA_M 900
#define LSA_N 50
#define LSA_THREADS 256

__global__ void __launch_bounds__(LSA_THREADS)
lsa_kernel(const float* __restrict__ C, float* __restrict__ outRows,
           float* __restrict__ outCols)
{
    const int b = blockIdx.x;
    const int t = threadIdx.x;
    const float* cost = C + (size_t)b * NQRY * T_TOT + b * T_EACH;

    __shared__ float u[LSA_N + 1];
    __shared__ float v[LSA_M + 1];
    __shared__ float minv[LSA_M + 1];
    __shared__ int   way[LSA_M + 1];
    __shared__ int   p[LSA_M + 1];
    __shared__ int   used[LSA_M + 1];
    __shared__ float red_val[LSA_THREADS];
    __shared__ int   red_idx[LSA_THREADS];
    __shared__ int   s_j0;
    __shared__ float s_delta;

    const float INF = __builtin_inff();

    for (int j = t; j <= LSA_M; j += LSA_THREADS) { v[j] = 0.0f; p[j] = 0; }
    for (int i = t; i <= LSA_N; i += LSA_THREADS)   u[i] = 0.0f;
    __syncthreads();

    for (int i = 1; i <= LSA_N; ++i) {
        if (t == 0) { p[0] = i; s_j0 = 0; }
        for (int j = t; j <= LSA_M; j += LSA_THREADS) { minv[j] = INF; used[j] = 0; }
        __syncthreads();

        for (;;) {
            const int j0 = s_j0;
            if (t == 0) used[j0] = 1;
            __syncthreads();

            const int   i0  = p[j0];
            const float ui0 = u[i0];

            // phase A: relax minv against row i0; local argmin over unused cols
            float bestv = INF; int bestj = LSA_M + 1;
            for (int j = 1 + t; j <= LSA_M; j += LSA_THREADS) {
                if (!used[j]) {
                    const float cur = cost[(size_t)(j - 1) * T_TOT + (i0 - 1)]
                                      - ui0 - v[j];
                    if (cur < minv[j]) { minv[j] = cur; way[j] = j0; }
                    const float mv = minv[j];
                    if (mv < bestv) { bestv = mv; bestj = j; }   // strided -> lowest j first
                }
            }
            red_val[t] = bestv; red_idx[t] = bestj;
            __syncthreads();
            // block argmin, lowest index on ties (matches np.argmin)
            for (int s = LSA_THREADS / 2; s > 0; s >>= 1) {
                if (t < s) {
                    const float ov = red_val[t + s]; const int oj = red_idx[t + s];
                    if (ov < red_val[t] || (ov == red_val[t] && oj < red_idx[t])) {
                        red_val[t] = ov; red_idx[t] = oj;
                    }
                }
                __syncthreads();
            }
            if (t == 0) { s_j0 = red_idx[0]; s_delta = red_val[0]; }
            __syncthreads();
            const int   j1    = s_j0;
            const float delta = s_delta;

            // phase C: update potentials
            for (int j = t; j <= LSA_M; j += LSA_THREADS) {
                if (used[j]) { u[p[j]] += delta; v[j] -= delta; }
                else         { minv[j] -= delta; }
            }
            __syncthreads();
            if (p[j1] == 0) break;   // shared, uniform across block
            __syncthreads();
        }

        // unwind augmenting path
        if (t == 0) {
            int j0 = s_j0;
            while (j0) { const int j1 = way[j0]; p[j0] = p[j1]; j0 = j1; }
        }
        __syncthreads();
    }

    // emit matched (query, target) pairs in ascending query order
    if (t == 0) {
        int k = 0;
        for (int j = 1; j <= LSA_M; ++j) {
            if (p[j] != 0) {
                outRows[b * T_EACH + k] = (float)(j - 1);      // query index
                outCols[b * T_EACH + k] = (float)(p[j] - 1);   // target index
                ++k;
            }
        }
    }
}

// =====================================================================
extern "C" void kernel_launch(void* const* d_in, const int* in_sizes, int n_in,
                              void* d_out, int out_size, void* d_ws, size_t ws_size,
                              hipStream_t stream)
{
    (void)in_sizes; (void)n_in; (void)out_size; (void)ws_size;
    const float* pred_logits = (const float*)d_in[0];  // [16,900,81]
    const float* pred_boxes  = (const float*)d_in[1];  // [16,900,4]
    const float* tgt_labels  = (const float*)d_in[2];  // [800,80]
    const float* tgt_boxes   = (const float*)d_in[3];  // [800,4]

    float* C       = (float*)d_out;                          // [16,900,800]
    float* outRows = C + (size_t)BSZ * NQRY * T_TOT;         // [16,50]
    float* outCols = outRows + BSZ * T_EACH;                 // [16,50]

    // workspace layout (floats), all 16B-aligned chunks
    float* d80   = (float*)d_ws;                 // [NQ][80]
    float* dlast = d80   + (size_t)NQ * KDIM;    // [NQ]
    float* pxyxy = dlast + NQ;                   // [NQ][4]
    float* parea = pxyxy + (size_t)NQ * 4;       // [NQ]
    float* txyxy = parea + NQ;                   // [T_TOT][4]
    float* tarea = txyxy + (size_t)T_TOT * 4;    // [T_TOT]

    // 1a) focal terms
    {
        const int total = NQ * NCLS;
        focal_kernel<<<(total + 255) / 256, 256, 0, stream>>>(pred_logits, d80, dlast);
    }
    // 1b) box corner/area precompute
    {
        const int total = NQ + T_TOT;
        box_prep_kernel<<<(total + 255) / 256, 256, 0, stream>>>(
            pred_boxes, tgt_boxes, pxyxy, parea, txyxy, tarea);
    }
    // 2) fused WMMA cost tiles: (900*50) tiles, 8 waves/block -> exact grid
    {
        const int tiles  = (NQ / 16) * (T_TOT / 16);   // 45000
        const int blocks = tiles / 8;                  // 5625 exactly
        cost_tile_kernel<<<blocks, 256, 0, stream>>>(d80, dlast, pred_boxes,
                                                     pxyxy, parea, tgt_labels,
                                                     tgt_boxes, txyxy, tarea, C);
    }
    // 3) per-batch Jonker-Volgenant assignment
    lsa_kernel<<<BSZ, LSA_THREADS, 0, stream>>>(C, outRows, outCols);
}